// MarginLoss_26938034881079
// MI455X (gfx1250) — compile-verified
//
#include <hip/hip_runtime.h>
#include <hip/hip_bf16.h>
#include <math.h>

// Problem constants (match reference)
#define B_      4096
#define K_      4
#define D_      512
#define T_      8192
#define NC_     100
#define MARGIN_ 0.2f
#define LAM_DIV_ 0.5f

typedef float        v2f   __attribute__((ext_vector_type(2)));
typedef float        v8f   __attribute__((ext_vector_type(8)));
typedef unsigned int u32x4 __attribute__((ext_vector_type(4)));
typedef unsigned int u32x8 __attribute__((ext_vector_type(8)));

__device__ __forceinline__ float waveReduceSum(float v) {
    // wave32: 5-step xor butterfly
    for (int off = 16; off > 0; off >>= 1)
        v += __shfl_xor(v, off, 32);
    return v;
}

// ---------------------------------------------------------------------------
// Zero the small accumulator region in workspace.
// acc layout: [0..K-1] per-group loss sum, [K..2K-1] per-group count,
//             [2K] diversity relu-sum (xs+xa combined)
// ---------------------------------------------------------------------------
__global__ void init_acc_kernel(float* acc) {
    if (threadIdx.x < 16) acc[threadIdx.x] = 0.0f;
}

// ---------------------------------------------------------------------------
// Per-row inverse L2 norms for the virtual concat [xs; x_augs] : [2B, K] f32.
// One wave per row; float4 (global_load_b128) coalesced reads.
// ---------------------------------------------------------------------------
__global__ void rownorm_kernel(const float* __restrict__ xs,
                               const float* __restrict__ xa,
                               float* __restrict__ invn) {
    int wave = (int)((blockIdx.x * blockDim.x + threadIdx.x) >> 5); // == row id
    int lane = threadIdx.x & 31;
    int i = wave / K_;
    int k = wave - i * K_;
    const float* row = (i < B_)
        ? xs + ((size_t)i * K_ + k) * D_
        : xa + ((size_t)(i - B_) * K_ + k) * D_;
    float s = 0.0f;
    for (int e = lane * 4; e < D_; e += 128) {
        float4 v = *(const float4*)(row + e);
        s += v.x * v.x + v.y * v.y + v.z * v.z + v.w * v.w;
    }
    s = waveReduceSum(s);
    if (lane == 0)
        invn[wave] = 1.0f / fmaxf(sqrtf(s), 1e-12f);
}

// ---------------------------------------------------------------------------
// Triplet margin loss: one wave per (k,t) triplet. Rows are gathered raw
// from xs/x_augs (L2-resident) and scaled by the precomputed inverse norms.
// float4 loads -> global_load_b128; per-block shared accumulation, then one
// float atomic per group per block.
// ---------------------------------------------------------------------------
__global__ void triplet_kernel(const float* __restrict__ xs,
                               const float* __restrict__ xa,
                               const float* __restrict__ beta,
                               const int*   __restrict__ ind,
                               const int*   __restrict__ trips,
                               const float* __restrict__ invn,
                               float* __restrict__ acc) {
    __shared__ float sSum[K_];
    __shared__ float sCnt[K_];
    if (threadIdx.x < K_) { sSum[threadIdx.x] = 0.0f; sCnt[threadIdx.x] = 0.0f; }
    __syncthreads();

    int lane = threadIdx.x & 31;
    int wave = (int)((blockIdx.x * blockDim.x + threadIdx.x) >> 5); // 0..K*T-1
    int k = wave / T_;
    int t = wave - k * T_;

    const int* tr = trips + ((size_t)k * T_ + t) * 3;
    int ai = tr[0], pi = tr[1], ni = tr[2];            // in [0, 2B)

    const float* Arow = (ai < B_) ? xs + ((size_t)ai * K_ + k) * D_
                                  : xa + ((size_t)(ai - B_) * K_ + k) * D_;
    const float* Prow = (pi < B_) ? xs + ((size_t)pi * K_ + k) * D_
                                  : xa + ((size_t)(pi - B_) * K_ + k) * D_;
    const float* Nrow = (ni < B_) ? xs + ((size_t)ni * K_ + k) * D_
                                  : xa + ((size_t)(ni - B_) * K_ + k) * D_;
    __builtin_prefetch(Prow + lane * 4, 0, 0);   // -> global_prefetch_b8
    __builtin_prefetch(Nrow + lane * 4, 0, 0);

    float ia  = invn[ai * K_ + k];
    float ip  = invn[pi * K_ + k];
    float inn = invn[ni * K_ + k];

    float dap = 0.0f, dan = 0.0f;
    for (int e = lane * 4; e < D_; e += 128) {
        float4 a4 = *(const float4*)(Arow + e);
        float4 p4 = *(const float4*)(Prow + e);
        float4 n4 = *(const float4*)(Nrow + e);
        float ax = a4.x * ia, ay = a4.y * ia, az = a4.z * ia, aw = a4.w * ia;
        float dp;
        dp = ax - p4.x * ip; dap += dp * dp;
        dp = ay - p4.y * ip; dap += dp * dp;
        dp = az - p4.z * ip; dap += dp * dp;
        dp = aw - p4.w * ip; dap += dp * dp;
        dp = ax - n4.x * inn; dan += dp * dp;
        dp = ay - n4.y * inn; dan += dp * dp;
        dp = az - n4.z * inn; dan += dp * dp;
        dp = aw - n4.w * inn; dan += dp * dp;
    }
    dap = waveReduceSum(dap);
    dan = waveReduceSum(dan);

    if (lane == 0) {
        int cls = ind[(ai < B_) ? ai : (ai - B_)];
        float b  = beta[k * NC_ + cls];
        float d_ap = sqrtf(dap + 1e-8f);
        float d_an = sqrtf(dan + 1e-8f);
        float pos = fmaxf(d_ap - b + MARGIN_, 0.0f);
        float neg = fmaxf(b - d_an + MARGIN_, 0.0f);
        atomicAdd(&sSum[k], pos + neg);
        if (pos > 0.0f || neg > 0.0f) atomicAdd(&sCnt[k], 1.0f);
    }
    __syncthreads();
    if (threadIdx.x < K_) {
        if (sSum[threadIdx.x] != 0.0f) atomicAdd(&acc[threadIdx.x],      sSum[threadIdx.x]);
        if (sCnt[threadIdx.x] != 0.0f) atomicAdd(&acc[K_ + threadIdx.x], sCnt[threadIdx.x]);
    }
}

// ---------------------------------------------------------------------------
// Diversity term: TDM + WMMA.
//
// The 16-row x 512-col tile (4 batch elems x K groups) is CONTIGUOUS in
// memory (rows are consecutive b*K+k indices), so one TENSOR_LOAD_TO_LDS
// moves the whole 32 KB tile. TDM pad_enable inserts 2 DWORDs every 256
// DWORDs -> LDS row stride 516 floats (516 % 64 = 4, so the 16 rows land in
// 16 distinct banks for column reads). Element (r,e) lives at
// r*516 + e + 2*(e>>8).
//
// Gram C[16x16] = (S·X)(S·X)^T is accumulated with V_WMMA_F32_16X16X4_F32
// over 128 K-chunks (full f32, matching the reference); row scaling by the
// inverse norms is folded into the fragment build (scale lane-row m by s[m]
// => C[M,N] scaled by s[M]*s[N]). For the symmetric product, the ISA's
// 16x4 A-layout and 4x16 B-layout give identical per-lane fragments:
// lane = {hi=lane[4], m=lane[3:0]}, frag = tile pair at (m, d0+2*hi).
// D layout: VGPR i holds (M=i+8*hi, N=m); keep ReLU of entries with M != N
// inside the same 4x4 diagonal block (same b, different group).
// ---------------------------------------------------------------------------
__global__ void div_wmma_kernel(const float* __restrict__ xs,
                                const float* __restrict__ xa,
                                const float* __restrict__ invn,
                                float* __restrict__ divAcc) {
    __shared__ float tile[16 * 516 + 8];   // ~33 KB, TDM-padded layout
    const int lane = threadIdx.x;          // blockDim.x == 32 (one wave)
    const int b0   = blockIdx.x * 4;
    const int sel  = blockIdx.y;           // 0 -> xs, 1 -> x_augs
    const float* X = sel ? xa : xs;
    const int rowOff = sel ? B_ : 0;

    // ---- Tensor DMA descriptor (D#), groups 0 and 1; 2D-capable form ----
    const float* gsrc = X + (size_t)(b0 * K_) * D_;       // 16 contiguous rows
    unsigned long long gaddr = (unsigned long long)(uintptr_t)gsrc;
    unsigned int lds_off = (unsigned int)(uintptr_t)&tile[0]; // low 32b = LDS offset

    u32x4 g0;
    g0[0] = 1u;                                            // count=1, load, no gather
    g0[1] = lds_off;                                       // lds_addr
    g0[2] = (unsigned int)(gaddr & 0xFFFFFFFFu);           // global_addr lo
    g0[3] = (unsigned int)((gaddr >> 32) & 0x01FFFFFFu)    // global_addr hi
          | (2u << 30);                                    // type = 2 ("image")

    u32x8 g1;
    g1[0] = (2u << 16)      // data_size = 4 bytes
          | (1u << 20)      // pad_enable
          | (7u << 22)      // pad_interval code 7 = every 256 DWORDs
          | (1u << 25);     // pad_amount  code 1 = 2 DWORDs
    g1[1] = (8192u & 0xFFFFu) << 16;   // tensor_dim0[15:0] = 8192 elements
    g1[2] = (8192u >> 16) | (1u << 16);// tensor_dim0 hi, tensor_dim1 = 1
    g1[3] = (8192u << 16);             // tensor_dim1 hi = 0, tile_dim0 = 8192
    g1[4] = 1u;                        // tile_dim1 = 1, tile_dim2 = 0
    g1[5] = 8192u;                     // tensor_dim0_stride lo
    g1[6] = 0u;                        // stride hi, tensor_dim1_stride lo
    g1[7] = 0u;

    // TDM: one DMA moves the whole tile into LDS (TENSORcnt-tracked).
    asm volatile("tensor_load_to_lds %0, %1" :: "s"(g0), "s"(g1) : "memory");
    __builtin_amdgcn_s_wait_tensorcnt(0);
    __syncthreads();   // single-wave workgroup: effectively free

    const int m  = lane & 15;
    const int hi = lane >> 4;
    const float sc = invn[(rowOff + b0 + (m >> 2)) * K_ + (m & 3)];
    const int rowBase = m * 516;

    v8f c = {};
    for (int d0 = 0; d0 < D_; d0 += 4) {
        int idx = d0 + 2 * hi;
        int off = rowBase + idx + 2 * (idx >> 8);      // TDM-padded address
        v2f a = *(const v2f*)&tile[off];               // 8B-aligned ds read
        a.x *= sc;
        a.y *= sc;
        // D = A (16x4) * A^T (4x16) + C, all f32
        c = __builtin_amdgcn_wmma_f32_16x16x4_f32(
                false, a, false, a, (short)0, c, false, false);
    }

    // Extract off-diagonal entries of the four 4x4 diagonal blocks, ReLU, sum.
    float s = 0.0f;
    for (int i = 0; i < 8; ++i) {
        int M = i + 8 * hi;   // C/D layout: VGPR i, lanes16-31 -> M=i+8
        int N = m;
        if ((M >> 2) == (N >> 2) && M != N)
            s += fmaxf(c[i], 0.0f);
    }
    s = waveReduceSum(s);
    if (lane == 0)
        atomicAdd(divAcc, s);
}

// ---------------------------------------------------------------------------
// Final scalar reduction -> (loss, loss_emb, loss_div, loss_global)
// ---------------------------------------------------------------------------
__global__ void finalize_kernel(const float* __restrict__ acc,
                                float* __restrict__ out) {
    if (threadIdx.x == 0 && blockIdx.x == 0) {
        float emb = 0.0f;
        for (int k = 0; k < K_; ++k)
            emb += acc[k] / fmaxf(acc[K_ + k], 1.0f);
        emb *= (1.0f / (float)K_);
        // 0.5*(S_xs/N + S_xa/N) with N = B*K*(K-1); both sums combined in acc[2K]
        float div = 0.5f * acc[2 * K_] / ((float)B_ * (float)K_ * (float)(K_ - 1));
        out[0] = emb + LAM_DIV_ * div;
        out[1] = emb;
        out[2] = div;
        out[3] = 0.0f;
    }
}

// ---------------------------------------------------------------------------
extern "C" void kernel_launch(void* const* d_in, const int* in_sizes, int n_in,
                              void* d_out, int out_size, void* d_ws, size_t ws_size,
                              hipStream_t stream) {
    const float* xs    = (const float*)d_in[0];   // [B,K,D] f32
    const float* xa    = (const float*)d_in[1];   // [B,K,D] f32
    const float* beta  = (const float*)d_in[2];   // [K,NC] f32
    const int*   ind   = (const int*)  d_in[3];   // [B] i32
    const int*   trips = (const int*)  d_in[4];   // [K,T,3] i32
    float*       out   = (float*)d_out;           // 4 scalars

    // workspace: invn [2B*K] f32 (128 KB) + 16-float accumulator block
    float* invn = (float*)d_ws;
    float* acc  = invn + 2 * B_ * K_;

    init_acc_kernel<<<1, 32, 0, stream>>>(acc);

    // 2*B*K = 32768 rows, 8 waves/block
    rownorm_kernel<<<(2 * B_ * K_) / 8, 256, 0, stream>>>(xs, xa, invn);

    // K*T = 32768 triplets, one wave each, 8 waves/block
    triplet_kernel<<<(K_ * T_) / 8, 256, 0, stream>>>(xs, xa, beta, ind, trips,
                                                      invn, acc);

    // B/4 tiles x {xs, xa}
    dim3 dgrid(B_ / 4, 2);
    div_wmma_kernel<<<dgrid, 32, 0, stream>>>(xs, xa, invn, acc + 2 * K_);

    finalize_kernel<<<1, 32, 0, stream>>>(acc, out);
}